// WAVE_16698832847495
// MI455X (gfx1250) — compile-verified
//
#include <hip/hip_runtime.h>
#include <math.h>

#define NSN 545          // N_NODES
#define DD  64
#define NB  2
#define HSZ (DD*NSN)     // 34880 floats per graph

typedef float v2f __attribute__((ext_vector_type(2)));
typedef float v8f __attribute__((ext_vector_type(8)));

// ---------------- layout transforms ----------------
__global__ void k_init(const float* __restrict__ x, float* __restrict__ H, int total) {
  int idx = blockIdx.x*blockDim.x + threadIdx.x;
  if (idx >= total) return;
  int b = idx / (NSN*DD);
  int r = idx % (NSN*DD);
  int node = r / DD, d = r % DD;
  H[b*HSZ + d*NSN + node] = x[idx];
}

__global__ void k_final(const float* __restrict__ H, float* __restrict__ out, int total) {
  int idx = blockIdx.x*blockDim.x + threadIdx.x;
  if (idx >= total) return;
  int b = idx / (NSN*DD);
  int r = idx % (NSN*DD);
  int node = r / DD, d = r % DD;
  out[idx] = H[b*HSZ + d*NSN + node];
}

__global__ void k_zero(float* __restrict__ p, int total) {
  int idx = blockIdx.x*blockDim.x + threadIdx.x;
  if (idx < total) p[idx] = 0.f;
}

__global__ void k_apply(float* __restrict__ H, const float* __restrict__ UPD, int total) {
  int idx = blockIdx.x*blockDim.x + threadIdx.x;
  if (idx < total) H[idx] += UPD[idx];
}

// ---------------- phase A: 8 projections, WMMA f32 16x16x4 ----------------
// Y(64 x m) = W(64x64 half) @ X(64 x m); one wave per 16x16 output tile.
// Out-of-range columns are address-clamped (not zeroed): they only pollute
// output columns that are never stored -> no divergent branches in the K loop.
__global__ void k_gemm_pq(const float* __restrict__ H, float* __restrict__ PQ,
                          const float* __restrict__ w_ta, const float* __restrict__ w_ca,
                          const float* __restrict__ w_tb, const float* __restrict__ w_cb,
                          int t, int c, int off, int icol,
                          int tilesT, int tilesC, int tilesPerB, int totalWaves)
{
  int wid  = (blockIdx.x*blockDim.x + threadIdx.x) >> 5;
  int lane = threadIdx.x & 31;
  if (wid >= totalWaves) return;
  int b  = wid / tilesPerB;
  int tw = wid % tilesPerB;
  int cnt[8] = {tilesT, tilesC, tilesC, tilesC, tilesT, tilesC, tilesC, tilesC};
  int mm = 0;
  for (; mm < 8; ++mm) { if (tw < cnt[mm]) break; tw -= cnt[mm]; }
  if (mm >= 8) return;
  int m      = (mm==0 || mm==4) ? t   : c;
  int coloff = (mm==0 || mm==4) ? off : icol;
  const float* Wsel;
  switch (mm) {
    case 0: Wsel = w_ta;      break;  // PT_a
    case 1: Wsel = w_ca;      break;  // PC_a
    case 2: Wsel = w_ta + 64; break;  // Q_ta
    case 3: Wsel = w_ca + 64; break;  // Q_ca
    case 4: Wsel = w_tb;      break;  // PT_b
    case 5: Wsel = w_cb;      break;  // PC_b
    case 6: Wsel = w_tb + 64; break;  // Q_tb
    default: Wsel = w_cb + 64;        // Q_cb
  }
  int ctiles = (m + 15) >> 4;
  int dt = tw / ctiles, nt = tw % ctiles;
  int d0 = dt*16, n0 = nt*16;
  const float* Xb = H + (size_t)b*HSZ + coloff;         // X[e][col] = Xb[e*NSN+col]
  float* Y = PQ + (size_t)(mm*NB + b)*HSZ;

  int mrow = lane & 15;
  int kb   = (lane >> 4) * 2;
  int coln = n0 + mrow;
  bool colok = coln < m;
  int colc  = colok ? coln : (m - 1);           // address clamp only
  const float* pA = Wsel + (d0 + mrow)*128 + kb;
  const float* pB = Xb   + (size_t)kb*NSN + colc;
  v8f acc = {0.f,0.f,0.f,0.f,0.f,0.f,0.f,0.f};
  #pragma unroll
  for (int k = 0; k < 64; k += 4) {
    v2f a, bb;
    a.x  = pA[k];
    a.y  = pA[k + 1];
    bb.x = pB[(size_t)k*NSN];
    bb.y = pB[(size_t)(k + 1)*NSN];
    acc = __builtin_amdgcn_wmma_f32_16x16x4_f32(false, a, false, bb, (short)0, acc, false, false);
  }
  int radd = (lane >> 4) * 8;
  if (colok) {
    #pragma unroll
    for (int r = 0; r < 8; ++r)
      Y[(d0 + r + radd)*NSN + coln] = acc[r];
  }
}

// ---------------- phase A2: per-d softmax factorization stats ----------------
__global__ void k_red(const float* __restrict__ H, const float* __restrict__ PQ,
                      float* __restrict__ RED, int t, int c, int off, int icol)
{
  int b = blockIdx.x;
  int d = threadIdx.x;          // 64 threads
  const float* PTA = PQ + (size_t)(0*NB + b)*HSZ + d*NSN;
  const float* PCA = PQ + (size_t)(1*NB + b)*HSZ + d*NSN;
  const float* Hb  = H  + (size_t)b*HSZ + d*NSN;
  float mt = -3.4e38f;
  for (int n = 0; n < t; ++n) mt = fmaxf(mt, PTA[n]);
  float st = 0.f, wt = 0.f;
  for (int n = 0; n < t; ++n) { float e = expf(PTA[n]-mt); st += e; wt += e*Hb[off+n]; }
  float mc = -3.4e38f;
  for (int n = 0; n < c; ++n) mc = fmaxf(mc, PCA[n]);
  float sc = 0.f, wc = 0.f;
  for (int n = 0; n < c; ++n) { float e = expf(PCA[n]-mc); sc += e; wc += e*Hb[icol+n]; }
  float* R = RED + b*384;
  R[d] = mt; R[64+d] = st; R[128+d] = wt; R[192+d] = mc; R[256+d] = sc; R[320+d] = wc;
}

// ---------------- phase B: m[k,d] = softmax part (factorized) + softsign part ----------------
#define NT 64
__global__ void k_phaseB(const float* __restrict__ H, const float* __restrict__ PQ,
                         const float* __restrict__ RED, float* __restrict__ MBUF,
                         const float* __restrict__ b_ta, const float* __restrict__ b_ca,
                         const float* __restrict__ b_tb, const float* __restrict__ b_cb,
                         const float* __restrict__ w_mix,
                         int t, int c, int off, int icol)
{
  __shared__ float sP[DD][NT+1];
  __shared__ float sN[DD][NT+1];
  int b   = blockIdx.y;
  int tid = threadIdx.x;             // 256 threads: 4 k x 64 d
  int kl  = tid >> 6;
  int d   = tid & 63;
  int k   = blockIdx.x*4 + kl;
  int kc  = (k < c) ? k : 0;
  const float* Hb  = H  + (size_t)b*HSZ;
  const float* QTA = PQ + (size_t)(2*NB+b)*HSZ;
  const float* QCA = PQ + (size_t)(3*NB+b)*HSZ;
  const float* PTB = PQ + (size_t)(4*NB+b)*HSZ;
  const float* PCB = PQ + (size_t)(5*NB+b)*HSZ;
  const float* QTB = PQ + (size_t)(6*NB+b)*HSZ;
  const float* QCB = PQ + (size_t)(7*NB+b)*HSZ;
  float qtb = QTB[d*NSN + kc] + b_tb[d];
  float qcb = QCB[d*NSN + kc] + b_cb[d];
  float accB = 0.f;
  // T block of the softsign sum
  for (int n0 = 0; n0 < t; n0 += NT) {
    __syncthreads();
    for (int xv = tid; xv < DD*NT; xv += 256) {
      int dd = xv >> 6, nn = xv & 63;
      int n  = n0 + nn;
      bool ok = n < t;
      int ncl = ok ? n : (t - 1);              // address clamp
      float pv = PTB[dd*NSN + ncl];
      float nv = Hb[dd*NSN + off + ncl];
      sP[dd][nn] = pv;                          // finite garbage OK (multiplied by 0)
      sN[dd][nn] = ok ? nv : 0.f;               // cndmask, no branch
    }
    __syncthreads();
    #pragma unroll 8
    for (int nn = 0; nn < NT; ++nn) {
      float p = sP[d][nn] + qtb;
      accB += (p / (1.f + fabsf(p))) * sN[d][nn];
    }
  }
  // C block of the softsign sum
  for (int n0 = 0; n0 < c; n0 += NT) {
    __syncthreads();
    for (int xv = tid; xv < DD*NT; xv += 256) {
      int dd = xv >> 6, nn = xv & 63;
      int n  = n0 + nn;
      bool ok = n < c;
      int ncl = ok ? n : (c - 1);
      float pv = PCB[dd*NSN + ncl];
      float nv = Hb[dd*NSN + icol + ncl];
      sP[dd][nn] = pv;
      sN[dd][nn] = ok ? nv : 0.f;
    }
    __syncthreads();
    #pragma unroll 8
    for (int nn = 0; nn < NT; ++nn) {
      float p = sP[d][nn] + qcb;
      accB += (p / (1.f + fabsf(p))) * sN[d][nn];
    }
  }
  // factorized softmax part
  const float* R = RED + b*384;
  float mt = R[d], st = R[64+d], wt = R[128+d];
  float mc = R[192+d], sc = R[256+d], wc = R[320+d];
  float aoff = QTA[d*NSN + kc] + b_ta[d] + mt;
  float coff = QCA[d*NSN + kc] + b_ca[d] + mc;
  float Mx = fmaxf(aoff, coff);
  float ea = expf(aoff - Mx), ec = expf(coff - Mx);
  float Z  = ea*st + ec*sc;
  float mA = w_mix[d] * (ea*wt + ec*wc) / Z;
  if (k < c) MBUF[(size_t)b*512*64 + k*64 + d] = mA + accB;
}

// ---------------- phase C: gates via fused K=128 WMMA GEMMs, write UPD ----------------
// Out-of-range k rows are address-clamped: they only pollute output rows
// that are never stored.
__global__ void k_phaseC(const float* __restrict__ H, const float* __restrict__ MBUF,
                         float* __restrict__ UPD,
                         const float* __restrict__ w_u, const float* __restrict__ b_u,
                         const float* __restrict__ w_o, const float* __restrict__ b_o,
                         int c, int icol, int ktiles, int totalWaves)
{
  int wid  = (blockIdx.x*blockDim.x + threadIdx.x) >> 5;
  int lane = threadIdx.x & 31;
  if (wid >= totalWaves) return;
  int b  = wid / (ktiles*4);
  int rr = wid % (ktiles*4);
  int kt = rr >> 2, dt = rr & 3;
  int k0 = kt*16, d0 = dt*16;
  int l15 = lane & 15;
  int kb  = (lane >> 4) * 2;
  int krow = k0 + l15;                   // A-matrix row (k)
  int dcol = d0 + l15;                   // B-matrix col (d)
  int krowc = (krow < c) ? krow : (c - 1);
  const float* Hb = H    + (size_t)b*HSZ;
  const float* Mb = MBUF + (size_t)b*512*64;
  const float* pM = Mb  + krowc*64 + kb;                     // K = 0..63  (m part)
  const float* pH = Hb  + (size_t)kb*NSN + icol + krowc;     // K = 64..127 (s = C^T part)
  const float* pU = w_u + dcol*128 + kb;
  const float* pO = w_o + dcol*128 + kb;
  v8f accU = {0.f,0.f,0.f,0.f,0.f,0.f,0.f,0.f};
  v8f accO = {0.f,0.f,0.f,0.f,0.f,0.f,0.f,0.f};
  #pragma unroll
  for (int e = 0; e < 128; e += 4) {
    v2f a, bu, bo;
    if (e < 64) {                        // compile-time uniform after unroll
      a.x = pM[e];
      a.y = pM[e + 1];
    } else {
      a.x = pH[(size_t)(e - 64)*NSN];
      a.y = pH[(size_t)(e - 63)*NSN];
    }
    bu.x = pU[e]; bu.y = pU[e + 1];
    bo.x = pO[e]; bo.y = pO[e + 1];
    accU = __builtin_amdgcn_wmma_f32_16x16x4_f32(false, a, false, bu, (short)0, accU, false, false);
    accO = __builtin_amdgcn_wmma_f32_16x16x4_f32(false, a, false, bo, (short)0, accO, false, false);
  }
  int radd = (lane >> 4) * 8;
  #pragma unroll
  for (int r = 0; r < 8; ++r) {
    int k = k0 + r + radd;
    int d = d0 + l15;
    if (k < c) {
      float u  = 1.f / (1.f + expf(-(accU[r] + b_u[d])));
      float o  = fmaxf(accO[r] + b_o[d], 0.f);
      float mv = Mb[k*64 + d];
      float s  = Hb[d*NSN + icol + k];
      UPD[(size_t)b*HSZ + d*NSN + icol + k] = u*o + (1.f - u)*mv - s;
    }
  }
}

// ---------------- host orchestration ----------------
extern "C" void kernel_launch(void* const* d_in, const int* in_sizes, int n_in,
                              void* d_out, int out_size, void* d_ws, size_t ws_size,
                              hipStream_t stream)
{
  (void)in_sizes; (void)n_in; (void)out_size; (void)ws_size;
  const float* x    = (const float*)d_in[0];
  // d_in[1] = edge_index: unused by the reference computation
  const float* w_ta = (const float*)d_in[2];
  const float* b_ta = (const float*)d_in[3];
  const float* w_ca = (const float*)d_in[4];
  const float* b_ca = (const float*)d_in[5];
  const float* w_tb = (const float*)d_in[6];
  const float* b_tb = (const float*)d_in[7];
  const float* w_cb = (const float*)d_in[8];
  const float* b_cb = (const float*)d_in[9];
  const float* w_mix= (const float*)d_in[10];
  const float* w_u  = (const float*)d_in[11];
  const float* b_u  = (const float*)d_in[12];
  const float* w_o  = (const float*)d_in[13];
  const float* b_o  = (const float*)d_in[14];
  float* out = (float*)d_out;

  float* H    = (float*)d_ws;
  float* UPD  = H    + NB*HSZ;
  float* MBUF = UPD  + NB*HSZ;
  float* PQ   = MBUF + NB*512*64;
  float* RED  = PQ   + 8*NB*HSZ;

  const int totalH  = NB*HSZ;
  const int blocksH = (totalH + 255)/256;
  k_init<<<blocksH,256,0,stream>>>(x, H, totalH);

  struct Seg { int t, c, off; };
  const Seg fwd[2] = {{1,32,0},   {32,512,1}};
  const Seg rev[2] = {{512,32,0}, {32,1,512}};

  for (int pass = 0; pass < 2; ++pass) {
    for (int dir = 0; dir < 2; ++dir) {
      const Seg* segs = (dir == 0) ? fwd : rev;
      k_zero<<<blocksH,256,0,stream>>>(UPD, totalH);
      for (int sgi = 0; sgi < 2; ++sgi) {
        int t = segs[sgi].t, c = segs[sgi].c, off = segs[sgi].off;
        int icol = off + t;
        int tilesT = 4*((t+15)/16);
        int tilesC = 4*((c+15)/16);
        int tilesPerB = 2*tilesT + 6*tilesC;
        int totalW = NB*tilesPerB;
        k_gemm_pq<<<(totalW+7)/8,256,0,stream>>>(H, PQ, w_ta, w_ca, w_tb, w_cb,
                                                 t, c, off, icol, tilesT, tilesC, tilesPerB, totalW);
        k_red<<<NB,64,0,stream>>>(H, PQ, RED, t, c, off, icol);
        dim3 gB((c+3)/4, NB);
        k_phaseB<<<gB,256,0,stream>>>(H, PQ, RED, MBUF, b_ta, b_ca, b_tb, b_cb, w_mix,
                                      t, c, off, icol);
        int ktiles  = (c+15)/16;
        int totalW2 = NB*ktiles*4;
        k_phaseC<<<(totalW2+7)/8,256,0,stream>>>(H, MBUF, UPD, w_u, b_u, w_o, b_o,
                                                 c, icol, ktiles, totalW2);
        k_apply<<<blocksH,256,0,stream>>>(H, UPD, totalH);
      }
    }
  }
  k_final<<<blocksH,256,0,stream>>>(H, out, totalH);
}